// Diffuse_88278757802358
// MI455X (gfx1250) — compile-verified
//
#include <hip/hip_runtime.h>

typedef __attribute__((ext_vector_type(2))) float v2f;
typedef __attribute__((ext_vector_type(8))) float v8f;

#define NB 4
#define NC 21
#define HW 4096
#define OC 64
#define IC 512
#define BN_EPS 1e-5f

// D(16x16 f32) = A(16x4 f32) * B(4x16 f32) + C    (wave32, CDNA5)
static __device__ __forceinline__ v8f wmma_f32(v2f a, v2f b, v8f c) {
    return __builtin_amdgcn_wmma_f32_16x16x4_f32(
        /*neg_a=*/false, a, /*neg_b=*/false, b,
        /*c_mod=*/(short)0, c, /*reuse_a=*/false, /*reuse_b=*/false);
}

// Sum across the 16 lanes of each half-wave via ds_swizzle (XOR patterns:
// offset = (xor_mask<<10) | and_mask(0x1f), group-of-32 mode).
static __device__ __forceinline__ float red16_add(float x) {
    x += __int_as_float(__builtin_amdgcn_ds_swizzle(__float_as_int(x), 0x041F));
    x += __int_as_float(__builtin_amdgcn_ds_swizzle(__float_as_int(x), 0x081F));
    x += __int_as_float(__builtin_amdgcn_ds_swizzle(__float_as_int(x), 0x101F));
    x += __int_as_float(__builtin_amdgcn_ds_swizzle(__float_as_int(x), 0x201F));
    return x;
}

// ---------------------------------------------------------------------------
// Kernel 0: pred_pad[b][cp][i] = cp < 21 ? pred[b][cp][i] : 0   (cp = 0..31)
// Lets kernel 3 load B-fragments unconditionally (N padded to 32).
// ---------------------------------------------------------------------------
__global__ void pad_pred_kernel(const float* __restrict__ pred,
                                float* __restrict__ pred_pad) {
    const int idx = blockIdx.x * 256 + threadIdx.x;    // over NB*32*HW
    const int i = idx & (HW - 1);
    const int c = (idx >> 12) & 31;
    const int b = idx >> 17;
    pred_pad[idx] = (c < NC) ? pred[((size_t)b * NC + c) * HW + i] : 0.0f;
}

// ---------------------------------------------------------------------------
// Kernel 1: Xt[b][i][o] = BN(conv1x1(feat)); also per-row norm^2 and per-batch
// max norm^2 (integer atomicMax is order-preserving for non-negative floats,
// and idempotent across graph replays).
// ---------------------------------------------------------------------------
__global__ void conv_bn_kernel(const float* __restrict__ feat,
                               const float* __restrict__ w,
                               const float* __restrict__ bias,
                               const float* __restrict__ gamma,
                               const float* __restrict__ betab,
                               const float* __restrict__ mean,
                               const float* __restrict__ var,
                               float* __restrict__ Xt,
                               float* __restrict__ norm2,
                               int*   __restrict__ maxn2) {
    __shared__ float part[4][16];                  // per-wave row partial sums
    const int b      = blockIdx.x >> 8;
    const int i0     = (blockIdx.x & 255) << 4;
    const int wv     = threadIdx.x >> 5;
    const int lane   = threadIdx.x & 31;
    const int lane16 = lane & 15;
    const int hi     = lane >> 4;          // 0: K=k,k+1  1: K=k+2,k+3
    const int o      = (wv << 4) + lane16; // output channel (N dim) per lane
    const size_t fb  = (size_t)b * IC * HW;
    const int arow   = i0 + lane16;        // A row (M dim) per lane

    v8f acc = {};
#pragma unroll 4
    for (int cc = 0; cc < IC; cc += 4) {
        const int ka = cc + hi * 2;
        v2f a, bb;
        a.x = feat[fb + (size_t)ka * HW + arow];
        a.y = feat[fb + (size_t)(ka + 1) * HW + arow];
        bb  = *(const v2f*)&w[o * IC + ka];
        acc = wmma_f32(a, bb, acc);
    }
    const float sc = gamma[o] * rsqrtf(var[o] + BN_EPS);
    const float sh = betab[o] - mean[o] * sc + bias[o] * sc;
#pragma unroll
    for (int r = 0; r < 8; ++r) {
        const int row   = i0 + r + hi * 8;         // D layout: rows r / r+8
        const float val = acc[r] * sc + sh;
        Xt[((size_t)b * HW + row) * OC + o] = val;
        const float q = red16_add(val * val);      // partial ||x_row||^2 (16 cols)
        if (lane16 == 0) part[wv][r + hi * 8] = q;
    }
    __syncthreads();
    if (threadIdx.x < 16) {
        const float t = part[0][threadIdx.x] + part[1][threadIdx.x] +
                        part[2][threadIdx.x] + part[3][threadIdx.x];
        norm2[b * HW + i0 + threadIdx.x] = t;
        atomicMax(&maxn2[b], __float_as_int(t));   // t >= 0: int order == float order
    }
}

// ---------------------------------------------------------------------------
// Kernel 2: single pass. S[i,j] = <x_i, x_j> via WMMA (K=64 -> 16 per tile),
// E = exp(S - bound_i) with bound_i = ||x_i|| * max||x|| >= max_j S[i,j]
// (softmax is shift-invariant; bound keeps exp in (0,1]). Each lane privately
// accumulates its column-residue share of the row sum; one cross-lane
// reduction at the very end. Store E into the P output region.
// ---------------------------------------------------------------------------
__global__ void gram_exp_kernel(const float* __restrict__ Xt,
                                const float* __restrict__ norm2,
                                const int*   __restrict__ maxn2,
                                float* __restrict__ E,
                                float* __restrict__ l_ws) {
    const int wv     = threadIdx.x >> 5;
    const int lane   = threadIdx.x & 31;
    const int lane16 = lane & 15;
    const int hi     = lane >> 4;
    const int iblk   = blockIdx.x * 4 + wv;
    const int b      = iblk >> 8;
    const int i0     = (iblk & 255) << 4;
    const size_t xb  = (size_t)b * HW * OC;
    const size_t pbase = (size_t)b * HW * HW;

    v2f afrag[16];                                 // A strip reused for all j
#pragma unroll
    for (int kk = 0; kk < 16; ++kk)
        afrag[kk] = *(const v2f*)&Xt[xb + (size_t)(i0 + lane16) * OC + kk * 4 + hi * 2];

    const float mn = sqrtf(__int_as_float(maxn2[b]));
    float bound[8], lsum[8];
#pragma unroll
    for (int r = 0; r < 8; ++r) {
        bound[r] = sqrtf(norm2[b * HW + i0 + r + hi * 8]) * mn;
        lsum[r] = 0.0f;
    }

    for (int jt = 0; jt < 256; ++jt) {
        const int j0 = jt << 4;
        v8f acc = {};
#pragma unroll
        for (int kk = 0; kk < 16; ++kk) {
            v2f bf = *(const v2f*)&Xt[xb + (size_t)(j0 + lane16) * OC + kk * 4 + hi * 2];
            acc = wmma_f32(afrag[kk], bf, acc);
        }
#pragma unroll
        for (int r = 0; r < 8; ++r) {              // hot loop: exp + add + store only
            const float e = __expf(acc[r] - bound[r]);
            lsum[r] += e;                          // lane-private partial row sum
            E[pbase + (size_t)(i0 + r + hi * 8) * HW + j0 + lane16] = e;
        }
    }
#pragma unroll
    for (int r = 0; r < 8; ++r) {                  // one cross-lane reduce, at the end
        const float t = red16_add(lsum[r]);
        if (lane16 == 0) l_ws[b * HW + i0 + r + hi * 8] = t;
    }
}

// ---------------------------------------------------------------------------
// Kernel 3: async-copy E tiles to LDS (GLOBAL_LOAD_ASYNC_TO_LDS_B128 +
// s_wait_asynccnt), write P = E/l back in place, and fuse out = (E*pred^T)/l
// via WMMA with A-fragments gathered from LDS. No exps, no __syncthreads:
// each wave's LDS region is private; per-wave counters order everything.
// LDS row stride 20 floats: 16B-aligned async chunks + conflict-free gathers.
// ---------------------------------------------------------------------------
__global__ void scale_out_kernel(float* __restrict__ P,        // in E, out probs
                                 const float* __restrict__ pred_pad,
                                 const float* __restrict__ pred,
                                 const float* __restrict__ seed,
                                 const float* __restrict__ l_ws,
                                 const float* __restrict__ alpha,
                                 const float* __restrict__ beta,
                                 float* __restrict__ out) {
    __shared__ float ldsE[4][16][20];
    const int wv     = threadIdx.x >> 5;
    const int lane   = threadIdx.x & 31;
    const int lane16 = lane & 15;
    const int hi     = lane >> 4;
    const int iblk   = blockIdx.x * 4 + wv;
    const int b      = iblk >> 8;
    const int i0     = (iblk & 255) << 4;
    const int lrow   = lane >> 1;                  // tile fetch: 2 lanes per row
    const int lcol4  = (lane & 1) << 2;            // 4-float chunks (B128)
    const float linvl = 1.0f / l_ws[b * HW + i0 + lrow];
    const size_t pbase   = (size_t)b * HW * HW;
    const size_t ppb     = (size_t)b * 32 * HW;    // padded pred base
    const size_t predb   = (size_t)b * NC * HW;
    const size_t rowbase = pbase + (size_t)(i0 + lrow) * HW + lcol4;
    const unsigned lds0  = (unsigned)(size_t)&ldsE[wv][lrow][lcol4];
    const int c1 = 16 + lane16;                    // second c-tile (21 -> pad 32)

    v8f oa0 = {}, oa1 = {};
    for (int jt = 0; jt < 256; ++jt) {
        const int j0 = jt << 4;
        const unsigned long long g0 = (unsigned long long)(size_t)&P[rowbase + j0];
        // previous iteration's ds-reads of this region have fully retired:
        asm volatile("s_wait_dscnt 0x0" ::: "memory");
        // INST_OFFSET applies to both LDS and global addresses (ISA 08 §4.4)
        asm volatile("global_load_async_to_lds_b128 %0, %1, off"
                     :: "v"(lds0), "v"(g0) : "memory");
        asm volatile("global_load_async_to_lds_b128 %0, %1, off offset:32"
                     :: "v"(lds0), "v"(g0) : "memory");
        asm volatile("s_wait_asynccnt 0x0" ::: "memory");

        // normalize the chunk this lane parked and write final P in place
        const float4 ea = *(const float4*)&ldsE[wv][lrow][lcol4];
        const float4 eb = *(const float4*)&ldsE[wv][lrow][lcol4 + 8];
        float4 pa = { ea.x * linvl, ea.y * linvl, ea.z * linvl, ea.w * linvl };
        float4 pv = { eb.x * linvl, eb.y * linvl, eb.z * linvl, eb.w * linvl };
        *(float4*)&P[rowbase + j0]     = pa;
        *(float4*)&P[rowbase + j0 + 8] = pv;

#pragma unroll
        for (int kk = 0; kk < 4; ++kk) {           // K=16 per tile -> 4 WMMA steps
            const int col = kk * 4 + hi * 2;
            const v2f af = *(const v2f*)&ldsE[wv][lane16][col];   // ds_load_b64
            const int j = j0 + col;                // B[k,c] = pred_pad[c][j]
            const v2f b0 = *(const v2f*)&pred_pad[ppb + (size_t)lane16 * HW + j];
            const v2f b1 = *(const v2f*)&pred_pad[ppb + (size_t)c1 * HW + j];
            oa0 = wmma_f32(af, b0, oa0);           // uniform flow: EXEC all-ones
            oa1 = wmma_f32(af, b1, oa1);
        }
    }

    const float av = alpha[0], bv = beta[0];
    const float sa_p = 1.0f / (1.0f + __expf(-av));   // sigmoid(alpha)
    const float sa_n = 1.0f - sa_p;                   // sigmoid(-alpha)
    const float sb_p = 1.0f / (1.0f + __expf(-bv));
    const float sb_n = 1.0f - sb_p;
#pragma unroll
    for (int r = 0; r < 8; ++r) {
        const int i = i0 + r + hi * 8;
        const float linv = 1.0f / l_ws[b * HW + i];
        const size_t idx0 = predb + (size_t)lane16 * HW + i;    // c = lane16 < 21
        out[idx0] = sb_n * (sa_n * (oa0[r] * linv) + sa_p * seed[idx0]) + sb_p * pred[idx0];
        if (c1 < NC) {
            const size_t idx1 = predb + (size_t)c1 * HW + i;
            out[idx1] = sb_n * (sa_n * (oa1[r] * linv) + sa_p * seed[idx1]) + sb_p * pred[idx1];
        }
    }
}

// ---------------------------------------------------------------------------
extern "C" void kernel_launch(void* const* d_in, const int* in_sizes, int n_in,
                              void* d_out, int out_size, void* d_ws, size_t ws_size,
                              hipStream_t stream) {
    const float* feat     = (const float*)d_in[0];
    const float* pred     = (const float*)d_in[1];
    const float* seed     = (const float*)d_in[2];
    const float* conv_w   = (const float*)d_in[3];
    const float* conv_b   = (const float*)d_in[4];
    const float* bn_gamma = (const float*)d_in[5];
    const float* bn_beta  = (const float*)d_in[6];
    const float* bn_mean  = (const float*)d_in[7];
    const float* bn_var   = (const float*)d_in[8];
    const float* alpha    = (const float*)d_in[9];
    const float* beta     = (const float*)d_in[10];

    float* out = (float*)d_out;
    float* P   = out + (size_t)NB * NC * HW;          // outputs: [out | P]

    float* Xt       = (float*)d_ws;                   // [NB][HW][OC]   4 MB
    float* l_ws     = Xt + (size_t)NB * HW * OC;      // [NB][HW]
    float* norm2    = l_ws + (size_t)NB * HW;         // [NB][HW]
    int*   maxn2    = (int*)(norm2 + (size_t)NB * HW);// [NB] (+pad to 16)
    float* pred_pad = (float*)(maxn2 + 16);           // [NB][32][HW]   2 MB

    pad_pred_kernel<<<dim3(NB * 32 * HW / 256), dim3(256), 0, stream>>>(pred, pred_pad);
    conv_bn_kernel<<<dim3(NB * 256), dim3(128), 0, stream>>>(
        feat, conv_w, conv_b, bn_gamma, bn_beta, bn_mean, bn_var, Xt, norm2, maxn2);
    gram_exp_kernel<<<dim3(NB * 64), dim3(128), 0, stream>>>(
        Xt, norm2, maxn2, P, l_ws);
    scale_out_kernel<<<dim3(NB * 64), dim3(128), 0, stream>>>(
        P, pred_pad, pred, seed, l_ws, alpha, beta, out);
}